// GateCappedSimplex_979252543919
// MI455X (gfx1250) — compile-verified
//
#include <hip/hip_runtime.h>

// GateCappedSimplex: row-wise projection onto {p : sum p = k, 0 <= p <= 1}
// via 30-step bisection. One wave32 per row; row resident in VGPRs.

#define ROW_LEN   2048
#define LANES     32
#define EPL       64            // elements per lane (ROW_LEN / LANES)
#define CHUNKS    16            // float4 chunks per lane (EPL / 4)
#define WPB       8             // waves per block
#define N_ITERS   30
#define BIG_NEG   -1000000000.0f

// ---- CDNA5 async global->LDS path selection -------------------------------
#if defined(__AMDGCN__)
  #if __has_builtin(__builtin_amdgcn_global_load_async_to_lds_b128)
    #define ASYNC_MODE 1        // clang builtin available
  #else
    #define ASYNC_MODE 2        // fall back to inline asm mnemonic
  #endif
#else
  #define ASYNC_MODE 0          // host pass: plain C++ only
#endif

#if ASYNC_MODE == 1
typedef __attribute__((ext_vector_type(4))) int v4i;
typedef __attribute__((address_space(1))) v4i GV4;   // global-AS int4
typedef __attribute__((address_space(3))) v4i LV4;   // LDS-AS int4
#endif

__device__ __forceinline__ void async_copy16(const float* gsrc, const float* ldst) {
#if ASYNC_MODE == 1
  GV4* gp = (GV4*)(uintptr_t)gsrc;                    // int->ptr, no addrspacecast
  LV4* lp = (LV4*)(unsigned)(uintptr_t)ldst;          // low 32 bits = LDS offset
  __builtin_amdgcn_global_load_async_to_lds_b128(gp, lp, 0, 0);
#elif ASYNC_MODE == 2
  unsigned           lds_off = (unsigned)(uintptr_t)ldst;
  unsigned long long gaddr   = (unsigned long long)(uintptr_t)gsrc;
  asm volatile("global_load_async_to_lds_b128 %0, %1, off"
               :: "v"(lds_off), "v"(gaddr) : "memory");
#endif
}

__device__ __forceinline__ void wait_async_zero() {
#if defined(__AMDGCN__)
  #if __has_builtin(__builtin_amdgcn_s_wait_asynccnt)
    __builtin_amdgcn_s_wait_asynccnt(0);
  #else
    asm volatile("s_wait_asynccnt 0" ::: "memory");
  #endif
  asm volatile("" ::: "memory");   // compiler barrier: don't hoist LDS reads
#endif
}

// ---- wave32 reductions (all lanes end with the reduced value) --------------
__device__ __forceinline__ float wave_sum_f32(float v) {
#pragma unroll
  for (int m = 16; m; m >>= 1) v += __shfl_xor(v, m, 32);
  return v;
}
__device__ __forceinline__ float wave_min_f32(float v) {
#pragma unroll
  for (int m = 16; m; m >>= 1) v = fminf(v, __shfl_xor(v, m, 32));
  return v;
}
__device__ __forceinline__ float wave_max_f32(float v) {
#pragma unroll
  for (int m = 16; m; m >>= 1) v = fmaxf(v, __shfl_xor(v, m, 32));
  return v;
}

__global__ __launch_bounds__(LANES * WPB, 1)
void GateCappedSimplex_979252543919_kernel(const float* __restrict__ s,
                                           const int*   __restrict__ mask,
                                           const int*   __restrict__ kptr,
                                           float*       __restrict__ out,
                                           int B) {
#if ASYNC_MODE
  __shared__ float smem[WPB * ROW_LEN];   // 64 KB: one row per wave
#endif
  const int lane = threadIdx.x & (LANES - 1);
  const int wave = threadIdx.x >> 5;
  const int row  = blockIdx.x * WPB + wave;
  if (row >= B) return;

  const float* srow = s + (size_t)row * ROW_LEN;
  const int4*  m4   = (const int4*)(mask + (size_t)row * ROW_LEN);

#if ASYNC_MODE
  float* lrow = &smem[wave * ROW_LEN];
#pragma unroll
  for (int c = 0; c < CHUNKS; ++c) {
    const int idx = c * LANES + lane;            // float4 index within row
    async_copy16(srow + idx * 4, lrow + idx * 4);
  }
#endif

  // Mask loads (plain vmem) overlap the in-flight async copies.
  int4 mreg[CHUNKS];
#pragma unroll
  for (int c = 0; c < CHUNKS; ++c) mreg[c] = m4[c * LANES + lane];

  const int   k     = *kptr;                     // scalar broadcast (s_load)
  const float k_eff = (float)(k < ROW_LEN ? k : ROW_LEN);

#if ASYNC_MODE
  wait_async_zero();
  const float4* s4 = (const float4*)lrow;        // ds_load_b128 from LDS
#else
  const float4* s4 = (const float4*)srow;        // direct global path
#endif

  // Build masked row in registers; fuse per-lane min/max.
  float v[EPL];
  float mn = 3.0e38f, mx = -3.0e38f;
#pragma unroll
  for (int c = 0; c < CHUNKS; ++c) {
    const float4 sv = s4[c * LANES + lane];
    const int4   mv = mreg[c];
    const float e0 = (mv.x > 0) ? sv.x : BIG_NEG;
    const float e1 = (mv.y > 0) ? sv.y : BIG_NEG;
    const float e2 = (mv.z > 0) ? sv.z : BIG_NEG;
    const float e3 = (mv.w > 0) ? sv.w : BIG_NEG;
    v[4 * c + 0] = e0; v[4 * c + 1] = e1;
    v[4 * c + 2] = e2; v[4 * c + 3] = e3;
    mn = fminf(mn, fminf(fminf(e0, e1), fminf(e2, e3)));
    mx = fmaxf(mx, fmaxf(fmaxf(e0, e1), fmaxf(e2, e3)));
  }

  float lo = wave_min_f32(mn) - 1.0f;
  float hi = wave_max_f32(mx);

  // 30-step bisection on threshold t; row stays in VGPRs, no barriers.
#pragma unroll 1
  for (int it = 0; it < N_ITERS; ++it) {
    const float t = 0.5f * (lo + hi);
    float a0 = 0.0f, a1 = 0.0f, a2 = 0.0f, a3 = 0.0f;
#pragma unroll
    for (int c = 0; c < CHUNKS; ++c) {
      a0 += fminf(fmaxf(v[4 * c + 0] - t, 0.0f), 1.0f);   // -> v_med3_f32
      a1 += fminf(fmaxf(v[4 * c + 1] - t, 0.0f), 1.0f);
      a2 += fminf(fmaxf(v[4 * c + 2] - t, 0.0f), 1.0f);
      a3 += fminf(fmaxf(v[4 * c + 3] - t, 0.0f), 1.0f);
    }
    const float g   = wave_sum_f32((a0 + a1) + (a2 + a3));
    const bool  big = g > k_eff;
    lo = big ? t : lo;
    hi = big ? hi : t;
  }

  const float t = 0.5f * (lo + hi);
  float4* o4 = (float4*)(out + (size_t)row * ROW_LEN);
#pragma unroll
  for (int c = 0; c < CHUNKS; ++c) {
    float4 o;
    o.x = fminf(fmaxf(v[4 * c + 0] - t, 0.0f), 1.0f);
    o.y = fminf(fmaxf(v[4 * c + 1] - t, 0.0f), 1.0f);
    o.z = fminf(fmaxf(v[4 * c + 2] - t, 0.0f), 1.0f);
    o.w = fminf(fmaxf(v[4 * c + 3] - t, 0.0f), 1.0f);
    o4[c * LANES + lane] = o;
  }
}

extern "C" void kernel_launch(void* const* d_in, const int* in_sizes, int n_in,
                              void* d_out, int out_size, void* d_ws, size_t ws_size,
                              hipStream_t stream) {
  const float* s    = (const float*)d_in[0];
  const int*   mask = (const int*)d_in[1];
  const int*   kptr = (const int*)d_in[2];
  float*       out  = (float*)d_out;
  (void)d_ws; (void)ws_size; (void)n_in; (void)out_size;

  const int B = in_sizes[0] / ROW_LEN;           // 16384 rows of 2048
  dim3 block(LANES * WPB);                       // 256 threads = 8 wave32
  dim3 grid((B + WPB - 1) / WPB);                // 2048 blocks
  GateCappedSimplex_979252543919_kernel<<<grid, block, 0, stream>>>(s, mask, kptr, out, B);
}